// MDCALoss_24172075942190
// MI455X (gfx1250) — compile-verified
//
#include <hip/hip_runtime.h>

typedef __attribute__((ext_vector_type(2))) float v2f;
typedef __attribute__((ext_vector_type(8))) float v8f;

#define E_EXITS 4
#define BATCH 8192
#define CLASSES 1000
#define TILES 64                        // wave-slots of 16 classes (63 real, slot 63 idle)
#define SPLITS 32                       // batch splits for parallelism
#define ROWS_PER_SPLIT (BATCH / SPLITS) // 256

// ---------------------------------------------------------------------------
// Kernel 1: batch column-sum via V_WMMA_F32_16X16X4_F32 with A = ones.
// One wave handles 16 classes for one (exit, batch-split). Loads are fully
// branchless: class index clamped for addressing, validity only gates the
// final partial-sum store (kernel 2 never reads padded classes).
// ---------------------------------------------------------------------------
__global__ __launch_bounds__(256) void mdca_colsum_wmma(
    const float* __restrict__ logits, float* __restrict__ ws) {
  const int lane  = threadIdx.x & 31;
  const int wid   = threadIdx.x >> 5;
  const int tile  = blockIdx.x * 8 + wid;  // 0..63
  const int e     = blockIdx.y;            // 0..3
  const int split = blockIdx.z;            // 0..31

  // Slot 63 covers classes 1008..1023: nothing real, uniform wave exit.
  if (tile * 16 >= CLASSES) return;

  const int  c_raw = tile * 16 + (lane & 15);      // class column this lane feeds
  const int  c     = c_raw < CLASSES ? c_raw : (CLASSES - 1);  // clamp for addressing
  const bool cv    = (c_raw < CLASSES);
  const int  rsel  = lane >> 4;                    // which row of each K-pair

  const size_t rowStride = (size_t)CLASSES;
  const float* base = logits + (size_t)e * BATCH * rowStride + (size_t)c;

  v2f a;
  a.x = 1.0f;
  a.y = 1.0f;  // A-matrix = ones (16x4), both K-VGPRs, all lanes

  v8f acc0 = {};
  v8f acc1 = {};

  const int b0row = split * ROWS_PER_SPLIT;
  const float* p = base + (size_t)b0row * rowStride + (size_t)rsel * rowStride;

#pragma unroll 4
  for (int it = 0; it < ROWS_PER_SPLIT; it += 8) {
    const float* r = p + (size_t)it * rowStride;
    v2f bm0, bm1;
    // B-matrix 4x16 f32: lanes 0..15 / 16..31 each carry one K-row per VGPR.
    // A is all-ones, so any bijective K assignment yields the same column sums.
    bm0.x = r[0 * rowStride];
    bm0.y = r[2 * rowStride];
    bm1.x = r[4 * rowStride];
    bm1.y = r[6 * rowStride];
    acc0 = __builtin_amdgcn_wmma_f32_16x16x4_f32(false, a, false, bm0,
                                                 (short)0, acc0, false, false);
    acc1 = __builtin_amdgcn_wmma_f32_16x16x4_f32(false, a, false, bm1,
                                                 (short)0, acc1, false, false);
  }

  v8f acc = acc0 + acc1;
  // D row 0: lanes 0..15 hold (M=0, N=lane) = column sums for classes tile*16+lane.
  if (lane < 16 && cv) {
    ws[(((size_t)e * TILES + tile) * SPLITS + split) * 16 + lane] = acc[0];
  }
}

// ---------------------------------------------------------------------------
// Kernel 2: histogram of target (deterministic integer LDS atomics), reduce
// the 32 split partials per (e, class), accumulate mean |avg_conf - avg_count|
// per exit with a shared-memory tree reduction. Writes 4 floats.
// ---------------------------------------------------------------------------
__global__ __launch_bounds__(256) void mdca_finalize(
    const float* __restrict__ ws, const long long* __restrict__ target,
    float* __restrict__ out) {
  __shared__ int   sh_cnt[TILES * 16];  // 1024 counters
  __shared__ float sh_red[E_EXITS][256];

  const int tid = threadIdx.x;

  for (int i = tid; i < TILES * 16; i += 256) sh_cnt[i] = 0;
  __syncthreads();

  for (int i = tid; i < BATCH; i += 256) {
    int t = (int)target[i];
    atomicAdd(&sh_cnt[t], 1);
  }
  __syncthreads();

  float local[E_EXITS] = {0.0f, 0.0f, 0.0f, 0.0f};
  for (int cc = tid; cc < CLASSES; cc += 256) {
    const float cnt  = (float)sh_cnt[cc];
    const int   tile = cc >> 4;
    const int   l    = cc & 15;
    for (int e = 0; e < E_EXITS; ++e) {
      const float* pp = ws + (((size_t)e * TILES + tile) * SPLITS) * 16 + l;
      float s = 0.0f;
#pragma unroll
      for (int sp = 0; sp < SPLITS; ++sp) s += pp[sp * 16];
      // |sum/B - cnt/B| = |sum - cnt| / B
      local[e] += fabsf((s - cnt) * (1.0f / (float)BATCH));
    }
  }

  for (int e = 0; e < E_EXITS; ++e) sh_red[e][tid] = local[e];
  __syncthreads();
  for (int off = 128; off > 0; off >>= 1) {
    if (tid < off) {
      for (int e = 0; e < E_EXITS; ++e) sh_red[e][tid] += sh_red[e][tid + off];
    }
    __syncthreads();
  }
  if (tid < E_EXITS) out[tid] = sh_red[tid][0] * (1.0f / (float)CLASSES);
}

// ---------------------------------------------------------------------------
// Launch: kernel1 streams the 131MB logits tensor (HBM-bound, ~5.6us floor),
// kernel2 is a single tiny block. Workspace use: 4*64*32*16 floats = 512 KB.
// ---------------------------------------------------------------------------
extern "C" void kernel_launch(void* const* d_in, const int* in_sizes, int n_in,
                              void* d_out, int out_size, void* d_ws,
                              size_t ws_size, hipStream_t stream) {
  (void)in_sizes;
  (void)n_in;
  (void)out_size;
  (void)ws_size;

  const float*     logits = (const float*)d_in[0];
  const long long* target = (const long long*)d_in[1];  // int64 targets
  float*           out    = (float*)d_out;
  float*           ws     = (float*)d_ws;

  dim3 grid1(TILES / 8, E_EXITS, SPLITS);  // 8 x 4 x 32 blocks, 8 waves each
  mdca_colsum_wmma<<<grid1, 256, 0, stream>>>(logits, ws);
  mdca_finalize<<<1, 256, 0, stream>>>(ws, target, out);
}